// HMM_71468255805641
// MI455X (gfx1250) — compile-verified
//
#include <hip/hip_runtime.h>
#include <math.h>

#define HMM_BATCH  8192
#define HMM_DIM    2048
#define HMM_CHUNKS 16
#define HMM_CLEN   (HMM_DIM / HMM_CHUNKS)   // 128 steps per chunk
#define HMM_WPC    (HMM_CLEN / 4)           // 32 int4 words per chunk

typedef __attribute__((ext_vector_type(2))) float v2f;
typedef __attribute__((ext_vector_type(8))) float v8f;

__device__ __forceinline__ void softmax3(const float* in, float* out) {
  float m  = fmaxf(in[0], fmaxf(in[1], in[2]));
  float e0 = expf(in[0] - m), e1 = expf(in[1] - m), e2 = expf(in[2] - m);
  float inv = 1.0f / (e0 + e1 + e2);
  out[0] = e0 * inv; out[1] = e1 * inv; out[2] = e2 * inv;
}

// ---------------------------------------------------------------------------
// Kernel 1: chunked scaled-forward scan.
//  chunk 0      : alpha row-vector (3) through steps t=0..127
//  chunks 1..15 : 3x3 matrix product of per-step M_{y_t}
// Linear domain, exact power-of-2 renorm every 4 steps (frexp/ldexp),
// exponent accumulated as int. No transcendentals in the hot loop.
// ---------------------------------------------------------------------------
__global__ void __launch_bounds__(256) hmm_scan(
    const int*   __restrict__ y,
    const float* __restrict__ Tlog,   // [3,3] transition logits
    const float* __restrict__ Elog,   // [3,2] emission logits
    const float* __restrict__ Slog,   // [3]   start logits
    float* __restrict__ wsP,          // [CHUNKS][9][BATCH]
    int*   __restrict__ wsE)          // [CHUNKS][BATCH]
{
  const int b = blockIdx.x * blockDim.x + threadIdx.x;
  const int c = blockIdx.y;

  // --- one-time parameter preparation (cheap, out of the hot loop) ---
  float T[9];
  #pragma unroll
  for (int i = 0; i < 3; ++i) softmax3(Tlog + 3 * i, T + 3 * i);

  float E0[3], E1[3];
  #pragma unroll
  for (int s = 0; s < 3; ++s) {
    float a = Elog[2 * s + 0], bb = Elog[2 * s + 1];
    float m = fmaxf(a, bb);
    float ea = expf(a - m), eb = expf(bb - m);
    float inv = 1.0f / (ea + eb);
    E0[s] = ea * inv; E1[s] = eb * inv;
  }
  float M0[9], M1[9];
  #pragma unroll
  for (int i = 0; i < 3; ++i)
    #pragma unroll
    for (int j = 0; j < 3; ++j) {
      M0[3 * i + j] = T[3 * i + j] * E0[j];
      M1[3 * i + j] = T[3 * i + j] * E1[j];
    }

  const int4* yp = reinterpret_cast<const int4*>(
      y + (size_t)b * HMM_DIM + (size_t)c * HMM_CLEN);
  int eacc = 0;

  if (c == 0) {
    // ---- vector chunk: alpha0 then 127 matvec steps ----
    float pi[3]; softmax3(Slog, pi);
    float v[3];
    int4 q = yp[0];
    #pragma unroll
    for (int s = 0; s < 3; ++s) v[s] = pi[s] * (q.x ? E1[s] : E0[s]);

    auto stepv = [&](int yv) {
      float Ms[9];
      #pragma unroll
      for (int k = 0; k < 9; ++k) Ms[k] = yv ? M1[k] : M0[k];
      float n0 = fmaf(v[0], Ms[0], fmaf(v[1], Ms[3], v[2] * Ms[6]));
      float n1 = fmaf(v[0], Ms[1], fmaf(v[1], Ms[4], v[2] * Ms[7]));
      float n2 = fmaf(v[0], Ms[2], fmaf(v[1], Ms[5], v[2] * Ms[8]));
      v[0] = n0; v[1] = n1; v[2] = n2;
    };
    auto renv = [&]() {
      float s = v[0] + v[1] + v[2];
      int ex; frexpf(s, &ex);
      float sc = ldexpf(1.0f, -ex);
      v[0] *= sc; v[1] *= sc; v[2] *= sc;
      eacc += ex;
    };

    stepv(q.y); stepv(q.z); stepv(q.w); renv();
    for (int w = 1; w < HMM_WPC; ++w) {
      if ((w & 7) == 0 && w + 8 < HMM_WPC)
        __builtin_prefetch(yp + w + 8, 0, 0);   // gfx1250 global_prefetch_b8
      int4 qq = yp[w];
      stepv(qq.x); stepv(qq.y); stepv(qq.z); stepv(qq.w);
      renv();
    }
    wsP[(0 * 9 + 0) * HMM_BATCH + b] = v[0];
    wsP[(0 * 9 + 1) * HMM_BATCH + b] = v[1];
    wsP[(0 * 9 + 2) * HMM_BATCH + b] = v[2];
    wsE[0 * HMM_BATCH + b] = eacc;
  } else {
    // ---- matrix chunk: P = prod of 128 per-step 3x3 matrices ----
    float P[9] = {1.f, 0.f, 0.f, 0.f, 1.f, 0.f, 0.f, 0.f, 1.f};
    auto stepP = [&](int yv) {
      float Ms[9];
      #pragma unroll
      for (int k = 0; k < 9; ++k) Ms[k] = yv ? M1[k] : M0[k];
      float N[9];
      #pragma unroll
      for (int i = 0; i < 3; ++i) {
        N[3*i+0] = fmaf(P[3*i+0], Ms[0], fmaf(P[3*i+1], Ms[3], P[3*i+2] * Ms[6]));
        N[3*i+1] = fmaf(P[3*i+0], Ms[1], fmaf(P[3*i+1], Ms[4], P[3*i+2] * Ms[7]));
        N[3*i+2] = fmaf(P[3*i+0], Ms[2], fmaf(P[3*i+1], Ms[5], P[3*i+2] * Ms[8]));
      }
      #pragma unroll
      for (int k = 0; k < 9; ++k) P[k] = N[k];
    };
    auto renP = [&]() {
      float s = (((P[0]+P[1]) + (P[2]+P[3])) + ((P[4]+P[5]) + (P[6]+P[7]))) + P[8];
      int ex; frexpf(s, &ex);
      float sc = ldexpf(1.0f, -ex);
      #pragma unroll
      for (int k = 0; k < 9; ++k) P[k] *= sc;
      eacc += ex;
    };

    for (int w = 0; w < HMM_WPC; ++w) {
      if ((w & 7) == 0 && w + 8 < HMM_WPC)
        __builtin_prefetch(yp + w + 8, 0, 0);
      int4 qq = yp[w];
      stepP(qq.x); stepP(qq.y); stepP(qq.z); stepP(qq.w);
      renP();
    }
    #pragma unroll
    for (int k = 0; k < 9; ++k) wsP[(c * 9 + k) * HMM_BATCH + b] = P[k];
    wsE[c * HMM_BATCH + b] = eacc;
  }
}

// ---------------------------------------------------------------------------
// Kernel 2: per-batch combine (chain 15 small matvecs) + block reduction.
// ---------------------------------------------------------------------------
__global__ void __launch_bounds__(256) hmm_combine(
    const float* __restrict__ wsP,
    const int*   __restrict__ wsE,
    float* __restrict__ wsPart)       // [gridDim.x] partial sums
{
  const int b = blockIdx.x * blockDim.x + threadIdx.x;
  float v0 = wsP[(0 * 9 + 0) * HMM_BATCH + b];
  float v1 = wsP[(0 * 9 + 1) * HMM_BATCH + b];
  float v2 = wsP[(0 * 9 + 2) * HMM_BATCH + b];
  int eacc = wsE[0 * HMM_BATCH + b];

  for (int c = 1; c < HMM_CHUNKS; ++c) {
    float p[9];
    #pragma unroll
    for (int k = 0; k < 9; ++k) p[k] = wsP[(c * 9 + k) * HMM_BATCH + b];
    eacc += wsE[c * HMM_BATCH + b];
    float n0 = fmaf(v0, p[0], fmaf(v1, p[3], v2 * p[6]));
    float n1 = fmaf(v0, p[1], fmaf(v1, p[4], v2 * p[7]));
    float n2 = fmaf(v0, p[2], fmaf(v1, p[5], v2 * p[8]));
    float s = n0 + n1 + n2;
    int ex; frexpf(s, &ex);
    float sc = ldexpf(1.0f, -ex);
    v0 = n0 * sc; v1 = n1 * sc; v2 = n2 * sc;
    eacc += ex;
  }
  float logZ = (float)eacc * 0.6931471805599453f + logf(v0 + v1 + v2);

  __shared__ float red[256];
  red[threadIdx.x] = logZ;
  __syncthreads();
  for (int off = 128; off > 0; off >>= 1) {
    if (threadIdx.x < off) red[threadIdx.x] += red[threadIdx.x + off];
    __syncthreads();
  }
  if (threadIdx.x == 0) wsPart[blockIdx.x] = red[0];
}

// ---------------------------------------------------------------------------
// Kernel 3: final deterministic reduction of 32 partials via f32 WMMA.
// A[16x4] holds the partials (K=0,1 used), B = ones -> column sums of D give
// the exact-order f32 total. One wave, EXEC all ones around the WMMA.
// ---------------------------------------------------------------------------
__global__ void hmm_final(const float* __restrict__ part, float* __restrict__ out)
{
  const int l = threadIdx.x;
  v2f a = {0.0f, 0.0f};
  if (l < 16) { a.x = part[2 * l + 0]; a.y = part[2 * l + 1]; }
  v2f bones = {1.0f, 1.0f};
  v8f cz = {0.f, 0.f, 0.f, 0.f, 0.f, 0.f, 0.f, 0.f};
  v8f d = __builtin_amdgcn_wmma_f32_16x16x4_f32(
      /*neg_a=*/false, a, /*neg_b=*/false, bones,
      /*c_mod=*/(short)0, cz, /*reuse_a=*/false, /*reuse_b=*/false);
  float loc = ((d[0] + d[1]) + (d[2] + d[3])) + ((d[4] + d[5]) + (d[6] + d[7]));
  float oth = __shfl_xor(loc, 16, 32);   // pair lanes (N, rows 0-7) with (N, rows 8-15)
  if (l == 0) out[0] = (loc + oth) * (1.0f / (float)HMM_BATCH);
}

extern "C" void kernel_launch(void* const* d_in, const int* in_sizes, int n_in,
                              void* d_out, int out_size, void* d_ws, size_t ws_size,
                              hipStream_t stream) {
  (void)in_sizes; (void)n_in; (void)out_size; (void)ws_size;
  const int*   y    = (const int*)d_in[0];
  const float* Tlog = (const float*)d_in[1];
  const float* Elog = (const float*)d_in[2];
  const float* Slog = (const float*)d_in[3];
  float* out = (float*)d_out;

  float* wsP    = (float*)d_ws;                                  // 16*9*8192 f32
  int*   wsE    = (int*)(wsP + (size_t)HMM_CHUNKS * 9 * HMM_BATCH); // 16*8192 i32
  float* wsPart = (float*)(wsE + (size_t)HMM_CHUNKS * HMM_BATCH);   // 32 f32

  dim3 scanGrid(HMM_BATCH / 256, HMM_CHUNKS);
  hmm_scan<<<scanGrid, 256, 0, stream>>>(y, Tlog, Elog, Slog, wsP, wsE);
  hmm_combine<<<dim3(HMM_BATCH / 256), 256, 0, stream>>>(wsP, wsE, wsPart);
  hmm_final<<<1, 32, 0, stream>>>(wsPart, out);
}